// SimpleAggregation3_57707180589696
// MI455X (gfx1250) — compile-verified
//
#include <hip/hip_runtime.h>
#include <hip/hip_bf16.h>

typedef float v2f __attribute__((ext_vector_type(2)));
typedef float v4f __attribute__((ext_vector_type(4)));
typedef float v8f __attribute__((ext_vector_type(8)));

#define B_DIM   64
#define N_DIM   8192
#define D_DIM   128
#define PCHUNK  32                 // N-chunks per batch
#define NROWS   (N_DIM / PCHUNK)   // 256 rows per chunk
#define WSTRIDE ((size_t)B_DIM * PCHUNK * D_DIM)  // one stat plane in ws

#define FLT_BIG 3.402823466e+38f

// ---------------- Stage 1: streaming partial reduction -------------------
// grid  = (PCHUNK, B_DIM), block = 256 (8 waves).
// Each wave consumes whole 128-float rows: lane owns d = lane*4 .. lane*4+3,
// loads are 128-bit non-temporal (dataset > L2, single use).
__global__ __launch_bounds__(256) void agg_stage1(const float* __restrict__ in,
                                                  float* __restrict__ ws) {
    const int p    = blockIdx.x;          // N-chunk
    const int b    = blockIdx.y;          // batch
    const int tid  = threadIdx.x;
    const int wave = tid >> 5;
    const int lane = tid & 31;

    // base of this (b, chunk) in elements; reinterpret as float4 stream
    const size_t base = ((size_t)b * N_DIM + (size_t)p * NROWS) * D_DIM;
    const v4f* ptr = (const v4f*)(in + base) + lane + (size_t)wave * (D_DIM / 4);

    v4f s  = {0.f, 0.f, 0.f, 0.f};
    v4f q  = {0.f, 0.f, 0.f, 0.f};
    v4f mx = {-FLT_BIG, -FLT_BIG, -FLT_BIG, -FLT_BIG};
    v4f mn = { FLT_BIG,  FLT_BIG,  FLT_BIG,  FLT_BIG};

    // 256 rows / 8 waves = 32 rows per wave, stride 8 rows
    #pragma unroll 4
    for (int r = 0; r < NROWS / 8; ++r) {
        v4f v = __builtin_nontemporal_load(ptr);   // global_load_b128, TH=NT
        ptr += 8 * (D_DIM / 4);
        s.x += v.x; s.y += v.y; s.z += v.z; s.w += v.w;
        q.x = __builtin_fmaf(v.x, v.x, q.x);
        q.y = __builtin_fmaf(v.y, v.y, q.y);
        q.z = __builtin_fmaf(v.z, v.z, q.z);
        q.w = __builtin_fmaf(v.w, v.w, q.w);
        mx.x = fmaxf(mx.x, v.x); mx.y = fmaxf(mx.y, v.y);
        mx.z = fmaxf(mx.z, v.z); mx.w = fmaxf(mx.w, v.w);
        mn.x = fminf(mn.x, v.x); mn.y = fminf(mn.y, v.y);
        mn.z = fminf(mn.z, v.z); mn.w = fminf(mn.w, v.w);
    }

    // cross-wave combine through LDS (8 waves -> 1 partial per block)
    __shared__ float lsum[8][D_DIM];
    __shared__ float lsq [8][D_DIM];
    __shared__ float lmx [8][D_DIM];
    __shared__ float lmn [8][D_DIM];
    const int d0 = lane * 4;
    lsum[wave][d0+0] = s.x;  lsum[wave][d0+1] = s.y;
    lsum[wave][d0+2] = s.z;  lsum[wave][d0+3] = s.w;
    lsq [wave][d0+0] = q.x;  lsq [wave][d0+1] = q.y;
    lsq [wave][d0+2] = q.z;  lsq [wave][d0+3] = q.w;
    lmx [wave][d0+0] = mx.x; lmx [wave][d0+1] = mx.y;
    lmx [wave][d0+2] = mx.z; lmx [wave][d0+3] = mx.w;
    lmn [wave][d0+0] = mn.x; lmn [wave][d0+1] = mn.y;
    lmn [wave][d0+2] = mn.z; lmn [wave][d0+3] = mn.w;
    __syncthreads();

    if (tid < D_DIM) {
        float S = 0.f, Q = 0.f, MX = -FLT_BIG, MN = FLT_BIG;
        #pragma unroll
        for (int w = 0; w < 8; ++w) {
            S += lsum[w][tid];
            Q += lsq [w][tid];
            MX = fmaxf(MX, lmx[w][tid]);
            MN = fminf(MN, lmn[w][tid]);
        }
        const size_t o = ((size_t)b * PCHUNK + p) * D_DIM + tid;
        ws[o]               = S;
        ws[WSTRIDE + o]     = Q;
        ws[2 * WSTRIDE + o] = MX;
        ws[3 * WSTRIDE + o] = MN;
    }
}

// ---------------- Stage 2: WMMA partial combine + finalize ----------------
// grid = (D_DIM/16, B_DIM), block = 32 (one wave).
// Sum and sum^2 partial reduction is a ones-vector matmul:
//   D = ones(16x4) x B(4x16) + C  via V_WMMA_F32_16X16X4_F32.
// With A == all-ones the K-slot permutation of the B layout is irrelevant,
// so the result is exact. Max/min use VALU + a lane-xor-16 shuffle.
__global__ __launch_bounds__(32) void agg_stage2(const float* __restrict__ ws,
                                                 float* __restrict__ out) {
    const int dc   = blockIdx.x;           // 16-wide d chunk
    const int b    = blockIdx.y;
    const int lane = threadIdx.x;
    const int nl   = lane & 15;            // column within chunk
    const int hi   = lane >> 4;            // half-wave id
    const int n    = dc * 16 + nl;         // d index

    const float* psum = ws + (size_t)b * PCHUNK * D_DIM;
    const float* psq  = psum + WSTRIDE;
    const float* pmx  = psum + 2 * WSTRIDE;
    const float* pmn  = psum + 3 * WSTRIDE;

    const v2f ones = {1.0f, 1.0f};
    v8f csum = {};
    v8f csq  = {};

    #pragma unroll
    for (int k0 = 0; k0 < PCHUNK; k0 += 4) {
        // B 4x16 tile: slot (vgpr v, half-wave hi) <- partial row k0 + 2v + hi
        const int ka = k0 + hi;            // goes in B vgpr 0
        const int kb = k0 + 2 + hi;        // goes in B vgpr 1
        v2f bs, bq;
        bs.x = psum[(size_t)ka * D_DIM + n];
        bs.y = psum[(size_t)kb * D_DIM + n];
        bq.x = psq [(size_t)ka * D_DIM + n];
        bq.y = psq [(size_t)kb * D_DIM + n];
        csum = __builtin_amdgcn_wmma_f32_16x16x4_f32(
            false, ones, false, bs, (short)0, csum, false, false);
        csq = __builtin_amdgcn_wmma_f32_16x16x4_f32(
            false, ones, false, bq, (short)0, csq, false, false);
    }

    // max/min: half-waves split the k range, then xor-16 combine
    float MX = -FLT_BIG, MN = FLT_BIG;
    for (int k = hi; k < PCHUNK; k += 2) {
        MX = fmaxf(MX, pmx[(size_t)k * D_DIM + n]);
        MN = fminf(MN, pmn[(size_t)k * D_DIM + n]);
    }
    MX = fmaxf(MX, __shfl_xor(MX, 16, 32));
    MN = fminf(MN, __shfl_xor(MN, 16, 32));

    if (lane < 16) {
        // D layout: vgpr0, lanes 0..15 hold row M=0, column N=lane
        const float S = csum[0];
        const float Q = csq[0];
        const float invN = 1.0f / (float)N_DIM;
        const float mean = S * invN;
        const float var  = fmaxf(Q * invN - mean * mean, 0.0f);
        const float sd   = __builtin_sqrtf(var);

        float* o = out + (size_t)b * (5 * D_DIM);
        o[n]             = mean;  // mean
        o[D_DIM + n]     = MX;    // max
        o[2 * D_DIM + n] = MN;    // min
        o[3 * D_DIM + n] = S;     // sum
        o[4 * D_DIM + n] = sd;    // std (population)
    }
}

extern "C" void kernel_launch(void* const* d_in, const int* in_sizes, int n_in,
                              void* d_out, int out_size, void* d_ws, size_t ws_size,
                              hipStream_t stream) {
    (void)in_sizes; (void)n_in; (void)out_size; (void)ws_size;
    const float* in = (const float*)d_in[0];
    float* out = (float*)d_out;
    float* ws  = (float*)d_ws;   // needs 4 * WSTRIDE * 4 B = 4 MB

    dim3 g1(PCHUNK, B_DIM);      // 2048 blocks x 256 threads: saturate HBM
    agg_stage1<<<g1, 256, 0, stream>>>(in, ws);

    dim3 g2(D_DIM / 16, B_DIM);  // 512 single-wave blocks
    agg_stage2<<<g2, 32, 0, stream>>>(ws, out);
}